// GNNLayer_5342939316511
// MI455X (gfx1250) — compile-verified
//
#include <hip/hip_runtime.h>
#include <hip/hip_bf16.h>

#define DIM 64

typedef __attribute__((ext_vector_type(2))) float v2f;
typedef __attribute__((ext_vector_type(8))) float v8f;

// ---------------------------------------------------------------------------
// 16-row x 64-col GEMM tile:  out[r, :] = A[r, :] @ W.T + b   using
// V_WMMA_F32_16X16X4_F32 (K-loop of 16 steps of K=4).
// A fragment (16x4 f32): lanes 0-15 -> row=lane, K=k0..k0+1 ; lanes 16-31 ->
// row=lane-16, K=k0+2..k0+3.  B fragment (4x16): col=lane&15, same K split.
// C/D (16x16 f32): col=lane&15, rows r + 8*(lane>=16).
// ---------------------------------------------------------------------------
__device__ __forceinline__ void gemm16_tile(const float (&a0)[16], const float (&a1)[16],
                                            const float* __restrict__ W,
                                            const float* __restrict__ bias,
                                            float* __restrict__ out,
                                            int row0, int nrows, int lane, bool full) {
  const int mrow  = lane & 15;
  const int khalf = (lane >> 4) << 1;   // 0 or 2
  const int rbase = (lane >> 4) * 8;    // 0 or 8
  for (int jt = 0; jt < 4; ++jt) {
    const int col = jt * 16 + mrow;
    const float* wr = W + (size_t)col * DIM;   // row `col` of W  ==  column of W.T
    v8f c = {};
#pragma unroll
    for (int ks = 0; ks < 16; ++ks) {
      v2f a; a.x = a0[ks]; a.y = a1[ks];
      v2f b; b.x = wr[ks * 4 + khalf]; b.y = wr[ks * 4 + khalf + 1];
      c = __builtin_amdgcn_wmma_f32_16x16x4_f32(false, a, false, b, (short)0, c, false, false);
    }
    const float bv = bias[col];
    float* obase = out + (size_t)(row0 + rbase) * DIM + col;
    if (full) {                            // uniform: unguarded coalesced stores
#pragma unroll
      for (int r = 0; r < 8; ++r) obase[(size_t)r * DIM] = c[r] + bv;
    } else {
#pragma unroll
      for (int r = 0; r < 8; ++r) {
        if (row0 + rbase + r < nrows) obase[(size_t)r * DIM] = c[r] + bv;
      }
    }
  }
}

__device__ __forceinline__ void load_a_tile(const float* __restrict__ src, int row0,
                                            int nrows, int lane,
                                            float (&a0)[16], float (&a1)[16]) {
  int arow = row0 + (lane & 15);
  if (arow >= nrows) arow = nrows - 1;
  const int khalf = (lane >> 4) << 1;
  const float* r = src + (size_t)arow * DIM;
#pragma unroll
  for (int ks = 0; ks < 16; ++ks) {
    a0[ks] = r[ks * 4 + khalf];
    a1[ks] = r[ks * 4 + khalf + 1];
  }
}

// --------------------------- kernels ---------------------------------------

__global__ void k_zero(float* __restrict__ p, long long n) {
  long long i = (long long)blockIdx.x * blockDim.x + threadIdx.x;
  if (i < n) p[i] = 0.0f;
}

// x1..x4 = x @ Wk.T + bk  (one wave per 16-row tile, WMMA)
__global__ void k_node_gemm(const float* __restrict__ x,
                            const float* __restrict__ W1, const float* __restrict__ b1,
                            const float* __restrict__ W2, const float* __restrict__ b2,
                            const float* __restrict__ W3, const float* __restrict__ b3,
                            const float* __restrict__ W4, const float* __restrict__ b4,
                            float* __restrict__ o1, float* __restrict__ o2,
                            float* __restrict__ o3, float* __restrict__ o4, int N) {
  const int lane = threadIdx.x & 31;
  const int wave = threadIdx.x >> 5;
  const int tile = blockIdx.x * (blockDim.x >> 5) + wave;
  const int row0 = tile * 16;
  if (row0 >= N) return;                  // wave-uniform: EXEC stays all-ones
  const bool full = (row0 + 16 <= N);     // wave-uniform
  float a0[16], a1[16];
  load_a_tile(x, row0, N, lane, a0, a1);
  gemm16_tile(a0, a1, W1, b1, o1, row0, N, lane, full);
  gemm16_tile(a0, a1, W2, b2, o2, row0, N, lane, full);
  gemm16_tile(a0, a1, W3, b3, o3, row0, N, lane, full);
  gemm16_tile(a0, a1, W4, b4, o4, row0, N, lane, full);
}

// sums[src] += sigmoid(w0) * x2[dst] ; counts[src] += 1
__global__ void k_edge_aggregate(const float* __restrict__ ea,
                                 const int* __restrict__ src, const int* __restrict__ dst,
                                 const float* __restrict__ x2,
                                 float* __restrict__ sums, float* __restrict__ counts,
                                 long long total) {
  long long idx = (long long)blockIdx.x * blockDim.x + threadIdx.x;
  if (idx >= total) return;
  const int e = (int)(idx >> 6);
  const int d = (int)(idx & 63);
  const int s = src[e], t = dst[e];
  const float w  = ea[idx];
  const float sw = 1.0f / (1.0f + __expf(-w));
  atomicAdd(&sums[(size_t)s * DIM + d], sw * x2[(size_t)t * DIM + d]);
  if (d == 0) atomicAdd(&counts[s], 1.0f);
}

// y = x1 + sums/max(cnt,1)  (in-place over sums) + per-feature BN stats
__global__ void k_node_combine_stats(const float* __restrict__ x1,
                                     float* __restrict__ sums,
                                     const float* __restrict__ counts,
                                     float* __restrict__ accum, long long total) {
  __shared__ float ssum[DIM], ssq[DIM];
  for (int i = threadIdx.x; i < DIM; i += blockDim.x) { ssum[i] = 0.f; ssq[i] = 0.f; }
  __syncthreads();
  long long idx = (long long)blockIdx.x * blockDim.x + threadIdx.x;
  if (idx < total) {
    const int n = (int)(idx >> 6), d = (int)(idx & 63);
    float cnt = counts[n]; if (cnt < 1.0f) cnt = 1.0f;
    const float y = x1[idx] + sums[idx] / cnt;
    sums[idx] = y;
    atomicAdd(&ssum[d], y);
    atomicAdd(&ssq[d], y * y);
  }
  __syncthreads();
  for (int i = threadIdx.x; i < DIM; i += blockDim.x) {
    atomicAdd(&accum[i], ssum[i]);
    atomicAdd(&accum[DIM + i], ssq[i]);
  }
}

// fold BN into z*scale+shift:  scale = gamma*rsqrt(var+eps), shift = beta - mean*scale
__global__ void k_bn_finalize(const float* __restrict__ accum,
                              const float* __restrict__ gamma, const float* __restrict__ beta,
                              float* __restrict__ ss, float recip) {
  const int d = threadIdx.x;
  if (d < DIM) {
    const float m  = accum[d] * recip;
    const float v  = accum[DIM + d] * recip - m * m;
    const float sc = rsqrtf(v + 1e-5f) * gamma[d];
    ss[d]        = sc;
    ss[DIM + d]  = beta[d] - m * sc;
  }
}

// x_out = x0 + silu(y*scale + shift)
__global__ void k_node_out(const float* __restrict__ x0, const float* __restrict__ y,
                           const float* __restrict__ ss, float* __restrict__ out,
                           long long total) {
  long long idx = (long long)blockIdx.x * blockDim.x + threadIdx.x;
  if (idx >= total) return;
  const int d = (int)(idx & 63);
  const float z = y[idx] * ss[d] + ss[DIM + d];
  out[idx] = x0[idx] + z / (1.0f + __expf(-z));
}

// Edge pass: t = w0@We.T + be + x3[src] + x4[dst].
// STATS=1: accumulate per-feature sum/sumsq.  STATS=0: w_out = w0 + silu(BN(t)).
template <int STATS>
__global__ void k_edge_pass(const float* __restrict__ ea,
                            const float* __restrict__ We, const float* __restrict__ be,
                            const int* __restrict__ srcI, const int* __restrict__ dstI,
                            const float* __restrict__ x3, const float* __restrict__ x4,
                            const float* __restrict__ ss, float* __restrict__ accum,
                            float* __restrict__ wout, int E) {
  __shared__ float ssum[DIM], ssq[DIM];
  if (STATS) {
    for (int i = threadIdx.x; i < DIM; i += blockDim.x) { ssum[i] = 0.f; ssq[i] = 0.f; }
    __syncthreads();
  }
  const int lane = threadIdx.x & 31;
  const int wave = threadIdx.x >> 5;
  const int tile = blockIdx.x * (blockDim.x >> 5) + wave;
  const int row0 = tile * 16;
  if (row0 < E) {                         // wave-uniform guard
    // speculative prefetch of a future A tile (dropped silently if OOB)
    __builtin_prefetch(ea + (size_t)(row0 + 16 * (blockDim.x >> 5)) * DIM + (lane << 3), 0, 1);
    const bool full = (row0 + 16 <= E);   // wave-uniform
    float a0[16], a1[16];
    load_a_tile(ea, row0, E, lane, a0, a1);
    const int mrow  = lane & 15;
    const int khalf = (lane >> 4) << 1;
    const int rbase = (lane >> 4) * 8;
    for (int jt = 0; jt < 4; ++jt) {
      const int col = jt * 16 + mrow;
      const float* wr = We + (size_t)col * DIM;
      v8f c = {};
#pragma unroll
      for (int ks = 0; ks < 16; ++ks) {
        v2f a; a.x = a0[ks]; a.y = a1[ks];
        v2f b; b.x = wr[ks * 4 + khalf]; b.y = wr[ks * 4 + khalf + 1];
        c = __builtin_amdgcn_wmma_f32_16x16x4_f32(false, a, false, b, (short)0, c, false, false);
      }
      const float bv = be[col];
#pragma unroll
      for (int r = 0; r < 8; ++r) {
        const int e = row0 + rbase + r;
        if (full || e < E) {
          const int s = srcI[e], t = dstI[e];
          const float z = c[r] + bv + x3[(size_t)s * DIM + col] + x4[(size_t)t * DIM + col];
          if (STATS) {
            atomicAdd(&ssum[col], z);
            atomicAdd(&ssq[col], z * z);
          } else {
            const float q = z * ss[col] + ss[DIM + col];
            wout[(size_t)e * DIM + col] = ea[(size_t)e * DIM + col] + q / (1.0f + __expf(-q));
          }
        }
      }
    }
  }
  if (STATS) {
    __syncthreads();
    for (int i = threadIdx.x; i < DIM; i += blockDim.x) {
      atomicAdd(&accum[i], ssum[i]);
      atomicAdd(&accum[DIM + i], ssq[i]);
    }
  }
}

// --------------------------- launcher ---------------------------------------

extern "C" void kernel_launch(void* const* d_in, const int* in_sizes, int n_in,
                              void* d_out, int out_size, void* d_ws, size_t ws_size,
                              hipStream_t stream) {
  const float* x  = (const float*)d_in[0];
  const float* ea = (const float*)d_in[1];
  const float* W1 = (const float*)d_in[2];  const float* b1 = (const float*)d_in[3];
  const float* W2 = (const float*)d_in[4];  const float* b2 = (const float*)d_in[5];
  const float* W3 = (const float*)d_in[6];  const float* b3 = (const float*)d_in[7];
  const float* W4 = (const float*)d_in[8];  const float* b4 = (const float*)d_in[9];
  const float* We = (const float*)d_in[10]; const float* be = (const float*)d_in[11];
  const float* v_gamma = (const float*)d_in[12]; const float* v_beta = (const float*)d_in[13];
  const float* e_gamma = (const float*)d_in[14]; const float* e_beta = (const float*)d_in[15];
  const int*   ei = (const int*)d_in[16];

  const int N = in_sizes[0] / DIM;
  const int E = in_sizes[1] / DIM;
  const int* srcI = ei;
  const int* dstI = ei + E;

  const size_t ND = (size_t)N * DIM;
  const size_t ED = (size_t)E * DIM;

  float* ws = (float*)d_ws;
  float* x1 = ws;
  float* x2 = ws + ND;
  float* x3 = ws + 2 * ND;
  float* x4 = ws + 3 * ND;
  float* sums   = ws + 4 * ND;       // becomes y after combine
  float* counts = ws + 5 * ND;       // N floats
  float* accumV = counts + N;        // 128
  float* accumE = accumV + 128;      // 128
  float* ssV    = accumE + 128;      // 128 (scale|shift)
  float* ssE    = ssV + 128;         // 128

  float* x_out = (float*)d_out;
  float* w_out = (float*)d_out + ND;

  const int TPB = 256;

  // 1) zero scratch (sums, counts, BN accumulators — contiguous)
  {
    long long nz = (long long)ND + N + 256;
    k_zero<<<(unsigned)((nz + TPB - 1) / TPB), TPB, 0, stream>>>(sums, nz);
  }

  // 2) node GEMMs (WMMA)
  {
    int tiles  = (N + 15) / 16;
    int blocks = (tiles + (TPB / 32) - 1) / (TPB / 32);
    k_node_gemm<<<blocks, TPB, 0, stream>>>(x, W1, b1, W2, b2, W3, b3, W4, b4,
                                            x1, x2, x3, x4, N);
  }

  // 3) gather + sigmoid gate + scatter-add
  {
    long long total = (long long)ED;
    k_edge_aggregate<<<(unsigned)((total + TPB - 1) / TPB), TPB, 0, stream>>>(
        ea, srcI, dstI, x2, sums, counts, total);
  }

  // 4) y = x1 + mean-agg ; node BN stats
  {
    long long total = (long long)ND;
    k_node_combine_stats<<<(unsigned)((total + TPB - 1) / TPB), TPB, 0, stream>>>(
        x1, sums, counts, accumV, total);
  }

  // 5) node BN fold
  k_bn_finalize<<<1, 64, 0, stream>>>(accumV, v_gamma, v_beta, ssV, 1.0f / (float)N);

  // 6) node output
  {
    long long total = (long long)ND;
    k_node_out<<<(unsigned)((total + TPB - 1) / TPB), TPB, 0, stream>>>(
        x, sums, ssV, x_out, total);
  }

  // 7) edge pass A: WMMA + gathers, BN stats only (recompute instead of spilling 400MB)
  {
    int tiles  = (E + 15) / 16;
    int blocks = (tiles + (TPB / 32) - 1) / (TPB / 32);
    k_edge_pass<1><<<blocks, TPB, 0, stream>>>(ea, We, be, srcI, dstI, x3, x4,
                                               ssE, accumE, w_out, E);
    // 8) edge BN fold
    k_bn_finalize<<<1, 64, 0, stream>>>(accumE, e_gamma, e_beta, ssE, 1.0f / (float)E);
    // 9) edge pass B: recompute + write w_out
    k_edge_pass<0><<<blocks, TPB, 0, stream>>>(ea, We, be, srcI, dstI, x3, x4,
                                               ssE, accumE, w_out, E);
  }
}